// GNN_72662256714256
// MI455X (gfx1250) — compile-verified
//
#include <hip/hip_runtime.h>

#define N_NODES 20000
#define KN      32
#define DFEAT   128
#define NODES_PER_BLOCK 32

typedef __attribute__((ext_vector_type(2))) float v2f;
typedef __attribute__((ext_vector_type(4))) float v4f;
typedef __attribute__((ext_vector_type(8))) float v8f;

// One block = 32 nodes, 256 threads = 8 waves.
// Phase 0: stage W (64KB) and the 32x32 adjacency slice into LDS.
// Phase 1: gather 32 neighbors per node with b128 loads, mean -> LDS.
// Phase 2: each wave owns one 16-col output tile and TWO 16-row A tiles:
//          two independent WMMA accumulator chains sharing each B fragment
//          (B via ds_load_2addr_b32 from LDS-resident W).
__global__ __launch_bounds__(256)
void gnn_layer_kernel(const float* __restrict__ h_prev,   // (N, D)
                      const int*   __restrict__ adj,      // (K, N)
                      const float* __restrict__ W,        // (D, D), W[d][e]
                      const float* __restrict__ bias,     // (D)
                      float*       __restrict__ h_out)    // (N, D)
{
    __shared__ float smW[DFEAT * DFEAT];            // 64KB: W staged row-major
    __shared__ float smA[NODES_PER_BLOCK * DFEAT];  // 16KB: mean-aggregated feats
    __shared__ int   smAdj[NODES_PER_BLOCK * KN];   //  4KB: neighbor ids

    const int tid = threadIdx.x;
    const int n0  = blockIdx.x * NODES_PER_BLOCK;

    // ---- Phase 0a: stage W into LDS (4096 float4, 16 per thread) ----
    {
        const v4f* __restrict__ W4 = reinterpret_cast<const v4f*>(W);
        v4f* __restrict__ smW4     = reinterpret_cast<v4f*>(smW);
        #pragma unroll
        for (int it = 0; it < 16; ++it)
            smW4[tid + it * 256] = W4[tid + it * 256];
    }

    // ---- Phase 0b: stage adjacency slice (1024 ints, 4 per thread) ----
    #pragma unroll
    for (int it = 0; it < 4; ++it) {
        int idx = tid + it * 256;                  // 0..1023
        int nl  = idx & (NODES_PER_BLOCK - 1);     // consecutive tids -> consec n
        int k   = idx >> 5;
        smAdj[nl * KN + k] = adj[k * N_NODES + n0 + nl];
    }
    __syncthreads();

    // ---- Phase 1: gather + mean over K neighbors into LDS (float4) ----
    // 32 nodes x 32 float4-chunks = 1024 chunks, 256 threads -> 4 each.
    const v4f* __restrict__ hp4 = reinterpret_cast<const v4f*>(h_prev);
    v4f* __restrict__ smA4      = reinterpret_cast<v4f*>(smA);

    #pragma unroll
    for (int it = 0; it < 4; ++it) {
        int idx = tid + it * 256;                  // 0..1023
        int nl  = idx >> 5;                        // local node 0..31
        int c4  = idx & 31;                        // float4 chunk within row
        v4f acc = {};
        #pragma unroll 8
        for (int k = 0; k < KN; ++k) {
            int nb = smAdj[nl * KN + k];           // LDS broadcast within wave
            acc += hp4[(size_t)nb * (DFEAT / 4) + c4];
        }
        smA4[nl * (DFEAT / 4) + c4] = acc * (1.0f / KN);
    }
    __syncthreads();

    // ---- Phase 2: WMMA GEMM (2 x 16 x 128) * (128 x 16) per wave ----
    const int wave    = tid >> 5;        // 0..7 -> output column tile
    const int lane    = tid & 31;
    const int half    = lane >> 4;       // ISA: lanes 16-31 hold K+2 / M+8
    const int l16     = lane & 15;
    const int colBase = wave * 16;

    v8f c0 = {};
    v8f c1 = {};
    #pragma unroll
    for (int ks = 0; ks < DFEAT / 4; ++ks) {
        const int k0 = ks * 4 + half * 2;
        // B fragment (4x16 f32): k = k0, k0+1, col = colBase + l16
        // (two LDS reads 512B apart -> single ds_load_2addr_b32)
        v2f bm;
        bm.x = smW[(k0    ) * DFEAT + colBase + l16];
        bm.y = smW[(k0 + 1) * DFEAT + colBase + l16];
        // A fragments for two row tiles (adjacent pair -> ds b64 loads)
        v2f a0, a1;
        a0.x = smA[(l16     ) * DFEAT + k0];
        a0.y = smA[(l16     ) * DFEAT + k0 + 1];
        a1.x = smA[(l16 + 16) * DFEAT + k0];
        a1.y = smA[(l16 + 16) * DFEAT + k0 + 1];
        // Two independent accumulator chains sharing bm
        c0 = __builtin_amdgcn_wmma_f32_16x16x4_f32(
                 false, a0, false, bm, (short)0, c0, false, false);
        c1 = __builtin_amdgcn_wmma_f32_16x16x4_f32(
                 false, a1, false, bm, (short)0, c1, false, false);
    }

    // ---- Epilogue: bias + relu + store (both row tiles) ----
    const int   col = colBase + l16;
    const float bv  = bias[col];
    #pragma unroll
    for (int r = 0; r < 8; ++r) {
        const int row = r + half * 8;              // C/D layout: M = r + half*8
        float v0 = c0[r] + bv;
        float v1 = c1[r] + bv;
        v0 = (v0 > 0.0f) ? v0 : 0.0f;
        v1 = (v1 > 0.0f) ? v1 : 0.0f;
        h_out[(size_t)(n0 + row     ) * DFEAT + col] = v0;
        h_out[(size_t)(n0 + row + 16) * DFEAT + col] = v1;
    }
}

extern "C" void kernel_launch(void* const* d_in, const int* in_sizes, int n_in,
                              void* d_out, int out_size, void* d_ws, size_t ws_size,
                              hipStream_t stream) {
    (void)in_sizes; (void)n_in; (void)out_size; (void)d_ws; (void)ws_size;

    const int*   adj   = (const int*)  d_in[0];   // (32, 20000) int32
    const float* graph = (const float*)d_in[1];   // (1, 20000, 128) f32
    const float* W     = (const float*)d_in[2];   // (3, 128, 128) f32
    const float* b     = (const float*)d_in[3];   // (3, 128) f32
    float*       out   = (float*)d_out;           // (1, 3, 20000, 128) f32

    const size_t layerElems = (size_t)N_NODES * DFEAT;

    // Layer 0: h0 = graph (device-to-device async copy is graph-capture safe)
    hipMemcpyAsync(out, graph, layerElems * sizeof(float),
                   hipMemcpyDeviceToDevice, stream);

    dim3 grid(N_NODES / NODES_PER_BLOCK);  // 625 blocks, exact
    dim3 block(256);                       // 8 waves of 32

    // Layer 1: reads graph, writes out[1]
    gnn_layer_kernel<<<grid, block, 0, stream>>>(
        graph, adj, W + 1 * DFEAT * DFEAT, b + 1 * DFEAT, out + 1 * layerElems);

    // Layer 2: reads out[1], writes out[2]
    gnn_layer_kernel<<<grid, block, 0, stream>>>(
        out + 1 * layerElems, adj, W + 2 * DFEAT * DFEAT, b + 2 * DFEAT,
        out + 2 * layerElems);
}